// EMAQuantizer_3186865733643
// MI455X (gfx1250) — compile-verified
//
#include <hip/hip_runtime.h>
#include <math.h>

// Problem constants (from reference): z [32,256,32,32], embedding [1024,256]
#define CDIM   256
#define KCODES 1024
#define NROWS  32768          // 32*32*32 flattened positions
#define ZP     17             // padded LDS stride: half-wave K-groups hit disjoint banks
#define TILEF  (CDIM * ZP)    // floats per staged 16x256 tile (4352)

typedef float v2f __attribute__((ext_vector_type(2)));
typedef float v8f __attribute__((ext_vector_type(8)));

// ---------------- Kernel 0: embedding norms + ws init ----------------
__global__ __launch_bounds__(256) void vq_init_kernel(const float* __restrict__ emb,
                                                      unsigned* __restrict__ counts,
                                                      float* __restrict__ enorm,
                                                      double* __restrict__ sumd) {
    int k = blockIdx.x * 256 + threadIdx.x;
    if (k < KCODES) {
        const float* row = emb + (size_t)k * CDIM;
        float s = 0.f;
        #pragma unroll 8
        for (int c = 0; c < CDIM; ++c) { float v = row[c]; s += v * v; }
        enorm[k]  = s;
        counts[k] = 0u;
    }
    if (k == 0) *sumd = 0.0;
}

// ---------------- Kernel 1: WMMA GEMM + argmin + gather ----------------
// Block: 128 threads (4 waves). Each block: 16 rows x all 1024 codes.
// Wave w owns column tiles [w*16, w*16+16), processed as 8 PAIRS so the
// A-operand LDS reads are shared by two WMMAs (3 ds-ops per 2 WMMAs).
__global__ __launch_bounds__(128) void vq_main_kernel(const float* __restrict__ z,
                                                      const float* __restrict__ emb,
                                                      const float* __restrict__ enorm,
                                                      unsigned* __restrict__ counts,
                                                      double* __restrict__ sumd,
                                                      float* __restrict__ zq) {
    // Single shared block: zs | es(4 waves x 2 tiles) | znorm | rowIdx | red
    // (argmin candidate buffers alias the es region after the GEMM loop)
    __shared__ __attribute__((aligned(16))) float smem[TILEF + 8 * TILEF + 16 + 16 + 128];
    float* zs      = smem;                       // zs[c*ZP + m]
    float* esall   = smem + TILEF;               // 8 tiles of 16x256, es[c*ZP + n]
    float* znorm_s = esall + 8 * TILEF;
    int*   rowIdx  = (int*)(znorm_s + 16);
    float* red     = (float*)(rowIdx + 16);
    float* cand_d  = esall;                      // alias: 128*8 floats
    int*   cand_i  = (int*)(esall + 1024);       // alias: 128*8 ints

    const int t    = threadIdx.x;
    const int wave = t >> 5;
    const int lane = t & 31;
    const int g    = lane >> 4;      // half-wave group (K/M split)
    const int q    = lane & 15;      // M index for A, N index for B/C/D
    const int kk   = g * 2;          // K offset within the 4-wide WMMA step

    const int rowbase = blockIdx.x * 16;
    const int bimg    = rowbase >> 10;       // batch index (HW = 1024)
    const int hw0     = rowbase & 1023;      // offset in 32x32 plane

    // ---- stage z strip: zs[c][m] = z[b][c][hw0+m] (coalesced 64B segments) ----
    for (int idx = t; idx < 16 * CDIM; idx += 128) {
        int m = idx & 15;
        int c = idx >> 4;
        zs[c * ZP + m] = z[((size_t)(bimg * CDIM + c) << 10) + hw0 + m];
    }
    __syncthreads();

    if (t < 16) {
        float s = 0.f;
        for (int c = 0; c < CDIM; ++c) { float v = zs[c * ZP + t]; s += v * v; }
        znorm_s[t] = s;
    }
    __syncthreads();

    float znr[8];
    #pragma unroll
    for (int j = 0; j < 8; ++j) znr[j] = znorm_s[g * 8 + j];   // M = g*8 + j

    float best[8];
    int   bestI[8];
    #pragma unroll
    for (int j = 0; j < 8; ++j) { best[j] = 3.0e38f; bestI[j] = 0; }
    float sum_d = 0.f;

    float* esw0 = esall + wave * (2 * TILEF);
    float* esw1 = esw0 + TILEF;

    for (int tt = 0; tt < 8; ++tt) {
        const int colbase0 = (wave * 16 + tt * 2) * 16;   // tile pair: cols, cols+16

        // stage 2 x (16x256) embedding tiles (float4 global loads -> es[c][n])
        #pragma unroll 4
        for (int i = 0; i < 64; ++i) {
            int f    = (lane + 32 * i) * 4;   // 0..8191
            int tile = f >> 12;               // 0 or 1
            int f2   = f & 4095;
            int nn   = f2 >> 8;               // tile row 0..15
            int c0   = f2 & 255;
            const float4 vv =
                *(const float4*)(emb + (size_t)(colbase0 + tile * 16 + nn) * CDIM + c0);
            float* dst = esw0 + tile * TILEF;
            dst[(c0 + 0) * ZP + nn] = vv.x;
            dst[(c0 + 1) * ZP + nn] = vv.y;
            dst[(c0 + 2) * ZP + nn] = vv.z;
            dst[(c0 + 3) * ZP + nn] = vv.w;
        }
        __syncthreads();   // uniform trip count across all 4 waves

        // ---- 2 x 64 V_WMMA_F32_16X16X4_F32 sharing the A-operand loads ----
        v8f acc0 = {};
        v8f acc1 = {};
        #pragma unroll 8
        for (int c = 0; c < CDIM; c += 4) {
            v2f a, b0, b1;
            a.x  = zs[(c + kk)     * ZP + q];
            a.y  = zs[(c + kk + 1) * ZP + q];
            b0.x = esw0[(c + kk)     * ZP + q];
            b0.y = esw0[(c + kk + 1) * ZP + q];
            b1.x = esw1[(c + kk)     * ZP + q];
            b1.y = esw1[(c + kk + 1) * ZP + q];
            acc0 = __builtin_amdgcn_wmma_f32_16x16x4_f32(
                       false, a, false, b0, (short)0, acc0, false, false);
            acc1 = __builtin_amdgcn_wmma_f32_16x16x4_f32(
                       false, a, false, b1, (short)0, acc1, false, false);
        }

        // dist = ||z||^2 + ||E||^2 - 2*S ; running argmin.
        // tile0 cols < tile1 cols and strict '<' keeps lowest index (argmin tie rule)
        const int   col0 = colbase0 + q;
        const int   col1 = col0 + 16;
        const float en0  = enorm[col0];
        const float en1  = enorm[col1];
        #pragma unroll
        for (int j = 0; j < 8; ++j) {
            float d0 = znr[j] + en0 - 2.0f * acc0[j];
            float d1 = znr[j] + en1 - 2.0f * acc1[j];
            sum_d += d0 + d1;
            if (d0 < best[j]) { best[j] = d0; bestI[j] = col0; }
            if (d1 < best[j]) { best[j] = d1; bestI[j] = col1; }
        }
        __syncthreads();   // before next pair overwrites esw0/esw1 (also orders alias use)
    }

    // ---- export per-lane candidates (into dead es region), reduce per row ----
    #pragma unroll
    for (int j = 0; j < 8; ++j) {
        cand_d[(t << 3) + j] = best[j];
        cand_i[(t << 3) + j] = bestI[j];
    }
    red[t] = sum_d;
    __syncthreads();

    if (t < 16) {
        const int m  = t;
        const int gg = m >> 3, jj = m & 7;
        float bd = 3.0e38f;
        int   bi = 0x7fffffff;
        for (int w = 0; w < 4; ++w) {          // waves cover increasing col ranges
            for (int l = 0; l < 16; ++l) {
                int   src = w * 32 + gg * 16 + l;
                float d   = cand_d[(src << 3) + jj];
                int   i   = cand_i[(src << 3) + jj];
                if (d < bd || (d == bd && i < bi)) { bd = d; bi = i; }
            }
        }
        rowIdx[m] = bi;
        atomicAdd(&counts[bi], 1u);
    }
    if (t == 0) {
        double tot = 0.0;
        for (int i = 0; i < 128; ++i) tot += (double)red[i];
        atomicAdd(sumd, tot);                  // global_atomic_add_f64
    }
    __syncthreads();

    // ---- gather z_q (NCHW): out[b][c][hw0+m] = E[idx[m]][c] ----
    {
        const int m    = t & 15;
        const int cb   = t >> 4;               // 0..7
        const int ridx = rowIdx[m];
        const float* erow = emb + (size_t)ridx * CDIM;
        #pragma unroll 4
        for (int i = 0; i < 32; ++i) {
            int c = cb * 32 + i;
            zq[((size_t)(bimg * CDIM + c) << 10) + hw0 + m] = erow[c];
        }
    }
}

// ---------------- Kernel 2: perplexity + mean distance ----------------
__global__ __launch_bounds__(256) void vq_final_kernel(const unsigned* __restrict__ counts,
                                                       const double* __restrict__ sumd,
                                                       float* __restrict__ out) {
    __shared__ float sh[256];
    int t = threadIdx.x;
    float local = 0.f;
    for (int k = t; k < KCODES; k += 256) {
        float e = (float)counts[k] * (1.0f / (float)NROWS);
        local += e * logf(e + 1e-10f);
    }
    sh[t] = local;
    __syncthreads();
    for (int s = 128; s > 0; s >>= 1) {
        if (t < s) sh[t] += sh[t + s];
        __syncthreads();
    }
    if (t == 0) {
        out[8388608] = expf(-sh[0]);                                     // perplexity
        out[8388609] = (float)(*sumd / (double)((size_t)NROWS * KCODES)); // mean dist
    }
}

extern "C" void kernel_launch(void* const* d_in, const int* in_sizes, int n_in,
                              void* d_out, int out_size, void* d_ws, size_t ws_size,
                              hipStream_t stream) {
    const float* z   = (const float*)d_in[0];   // [32,256,32,32]
    const float* emb = (const float*)d_in[1];   // [1024,256]
    float* out = (float*)d_out;                  // z_q (8388608) + perp + mean

    unsigned* counts = (unsigned*)d_ws;                       // 1024 u32
    float*    enorm  = (float*)d_ws + 1024;                   // 1024 f32
    double*   sumd   = (double*)((char*)d_ws + 8192);         // 1 f64 (8B aligned)

    vq_init_kernel <<<4,    256, 0, stream>>>(emb, counts, enorm, sumd);
    vq_main_kernel <<<NROWS / 16, 128, 0, stream>>>(z, emb, enorm, counts, sumd, out);
    vq_final_kernel<<<1,    256, 0, stream>>>(counts, sumd, out);
}